// OcclusionThirdLayer_36464272343263
// MI455X (gfx1250) — compile-verified
//
#include <hip/hip_runtime.h>
#include <stdint.h>

// OcclusionThirdLayer: out[r] = bias[r] + W[r,c1]*x[c1] + W[r,c2]*x[c2]
// with r = i*224 + j, c1 = i*448 + j, c2 = c1 + 224  (only positions assumed;
// W values and bias are read from the real inputs).
//
// Honest dense GEMV floor: 411 MB of W @ 23.3 TB/s ~= 17.6 us.
// Structured gather: ~115 KB useful traffic -> launch-latency bound.

#define SIZE_IN   14336
#define SIZE_OUT  7168
#define BS        224          // occlusion block size
#define NGROUPS   32           // SIZE_OUT / BS

#if defined(__AMDGCN__) && defined(__has_builtin)
#  if __has_builtin(__builtin_amdgcn_global_load_async_to_lds_b32)
#    define OCC_ASYNC_LDS 1
#  endif
#  if __has_builtin(__builtin_amdgcn_s_wait_asynccnt)
#    define OCC_HAVE_WAIT_BUILTIN 1
#  endif
#endif

#if defined(__AMDGCN__)
typedef __attribute__((address_space(1))) int* occ_gptr_t;  // global
typedef __attribute__((address_space(3))) int* occ_lptr_t;  // LDS
#endif

__global__ __launch_bounds__(BS) void occ3_kernel(const float* __restrict__ x,
                                                  const float* __restrict__ W,
                                                  const float* __restrict__ bias,
                                                  float* __restrict__ out) {
    __shared__ float xs[2 * BS];           // this group's 448-element slice of x

    const int i    = blockIdx.x;           // group 0..31
    const int j    = threadIdx.x;          // 0..223
    const int base = i * 2 * BS;           // start of group's x slice

#if OCC_ASYNC_LDS
    // CDNA5 async data mover: global -> LDS, tracked by ASYNCcnt, no VGPR data.
    __builtin_amdgcn_global_load_async_to_lds_b32(
        (occ_gptr_t)(x + base + j),
        (occ_lptr_t)(&xs[j]),
        /*offset=*/0, /*cpol=*/0);
    __builtin_amdgcn_global_load_async_to_lds_b32(
        (occ_gptr_t)(x + base + BS + j),
        (occ_lptr_t)(&xs[BS + j]),
        /*offset=*/0, /*cpol=*/0);
#  if OCC_HAVE_WAIT_BUILTIN
    __builtin_amdgcn_s_wait_asynccnt(0);
#  else
    asm volatile("s_wait_asynccnt 0x0" ::: "memory");
#  endif
#else
    xs[j]      = x[base + j];
    xs[BS + j] = x[base + BS + j];
#endif
    __syncthreads();

    const int      r    = i * BS + j;                    // output row
    const uint32_t wrow = (uint32_t)r * SIZE_IN;

    // The only two live coefficients of row r. Scattered, read-once -> NT hint.
    const float w1 = __builtin_nontemporal_load(W + wrow + base + j);
    const float w2 = __builtin_nontemporal_load(W + wrow + base + BS + j);

    out[r] = fmaf(w1, xs[j], fmaf(w2, xs[BS + j], bias[r]));
}

extern "C" void kernel_launch(void* const* d_in, const int* in_sizes, int n_in,
                              void* d_out, int out_size, void* d_ws, size_t ws_size,
                              hipStream_t stream) {
    (void)in_sizes; (void)n_in; (void)out_size; (void)d_ws; (void)ws_size;
    const float* x    = (const float*)d_in[0];
    const float* W    = (const float*)d_in[1];
    const float* bias = (const float*)d_in[2];
    float*       out  = (float*)d_out;

    occ3_kernel<<<dim3(NGROUPS), dim3(BS), 0, stream>>>(x, W, bias, out);
}